// AffinitySideLoss_1829656068254
// MI455X (gfx1250) — compile-verified
//
#include <hip/hip_runtime.h>

namespace {
constexpr int S = 8, B = 4, E = 12, H = 512, W = 512;
constexpr int HW  = H * W;
constexpr int EHW = E * HW;
constexpr float DELTA2     = 3.0f;        // 2 * DELTA
constexpr float INV_DELTA2 = 1.0f / 3.0f;
constexpr float EPS        = 1e-7f;
}

typedef float v2f __attribute__((ext_vector_type(2)));
typedef float v8f __attribute__((ext_vector_type(8)));

// Exact f32 sum over the 32 lanes of a wave, using V_WMMA_F32_16X16X4_F32.
// A = per-lane data (16x4 matrix; its 64 register slots are exactly the 32
// lane values + 32 zeros), B = all-ones 4x16 (layout-invariant). Then
// D[m,n] = rowsum_m for every n; each lane's 8 D-VGPRs hold 8 distinct row
// sums (rows 0-7 for lanes 0-15, rows 8-15 for lanes 16-31), so
// (sum of d[0..7]) + shfl_xor(.,16) == total, in every lane. All math is
// "x*1.0 + acc" in f32 -> bit-exactly a summation tree.
__device__ __forceinline__ float wave_sum32(float v) {
#if defined(__AMDGCN__) && __has_builtin(__builtin_amdgcn_wmma_f32_16x16x4_f32)
  v2f a;    a[0] = v;    a[1] = 0.0f;
  v2f ones; ones[0] = 1.0f; ones[1] = 1.0f;
  v8f c = {};
  v8f d = __builtin_amdgcn_wmma_f32_16x16x4_f32(
      /*neg_a=*/false, a, /*neg_b=*/false, ones,
      /*c_mod=*/(short)0, c, /*reuse_a=*/false, /*reuse_b=*/false);
  float h = ((d[0] + d[1]) + (d[2] + d[3])) + ((d[4] + d[5]) + (d[6] + d[7]));
  h += __shfl_xor(h, 16, 32);
  return h;
#else
  #pragma unroll
  for (int off = 16; off; off >>= 1) v += __shfl_xor(v, off, 32);
  return v;
#endif
}

__global__ void __launch_bounds__(256)
affinity_partials(const float* __restrict__ in, const int* __restrict__ tgt,
                  const int* __restrict__ offs, float* __restrict__ acc) {
  const int b = blockIdx.y;

  int oy[S], ox[S];
  #pragma unroll
  for (int s = 0; s < S; ++s) { oy[s] = offs[2 * s]; ox[s] = offs[2 * s + 1]; }

  float num[S], den[S], cnt[S];
  #pragma unroll
  for (int s = 0; s < S; ++s) { num[s] = 0.f; den[s] = 0.f; cnt[s] = 0.f; }

  const float* __restrict__ inb = in  + (size_t)b * EHW;
  const int*   __restrict__ tb  = tgt + (size_t)b * HW;

  const int stride = gridDim.x * blockDim.x;
  for (int p = blockIdx.x * blockDim.x + threadIdx.x; p < HW; p += stride) {
    const int i = p >> 9;          // W == 512
    const int j = p & (W - 1);

    float cv[E];                   // center embedding, loaded once, reused x8
    #pragma unroll
    for (int e = 0; e < E; ++e) cv[e] = inb[e * HW + p];
    const int tc = tb[p];

    #pragma unroll
    for (int s = 0; s < S; ++s) {
      int iy = i + oy[s]; iy = iy < 0 ? 0 : (iy > H - 1 ? H - 1 : iy);
      int ix = j + ox[s]; ix = ix < 0 ? 0 : (ix > W - 1 ? W - 1 : ix);
      const int q = iy * W + ix;

      float ss = 0.f;
      #pragma unroll
      for (int e = 0; e < E; ++e) {
        const float d = cv[e] - inb[e * HW + q];
        ss = fmaf(d, d, ss);
      }
      const float x   = fmaxf((DELTA2 - sqrtf(ss)) * INV_DELTA2, 0.f);
      const float aff = 1.f - x * x;              // affs (already inverted)
      const float t   = (tc == tb[q]) ? 0.f : 1.f; // target aff (inverted)

      num[s] = fmaf(aff, t,   num[s]);   // sum(affs * t)
      den[s] = fmaf(aff, aff, den[s]);   // sum(affs^2)
      cnt[s] += t;                       // sum(t^2) == sum(t), t is binary
    }
  }

  __shared__ float part[3 * S];
  if (threadIdx.x < 3 * S) part[threadIdx.x] = 0.f;
  __syncthreads();

  // Wave-level exact reduction via f32 WMMA, then LDS combine across waves.
  #pragma unroll
  for (int s = 0; s < S; ++s) {
    const float n = wave_sum32(num[s]);
    const float d = wave_sum32(den[s]);
    const float c = wave_sum32(cnt[s]);
    if ((threadIdx.x & 31) == 0) {
      atomicAdd(&part[3 * s + 0], n);
      atomicAdd(&part[3 * s + 1], d);
      atomicAdd(&part[3 * s + 2], c);
    }
  }
  __syncthreads();
  if (threadIdx.x < 3 * S) atomicAdd(&acc[threadIdx.x], part[threadIdx.x]);
}

__global__ void zero_acc(float* __restrict__ acc) {
  if (threadIdx.x < 3 * S) acc[threadIdx.x] = 0.f;
}

__global__ void finalize(const float* __restrict__ acc, float* __restrict__ out) {
  if (threadIdx.x == 0) {
    float tot = 0.f;
    #pragma unroll
    for (int s = 0; s < S; ++s) {
      const float n = acc[3 * s + 0];
      const float d = acc[3 * s + 1] + acc[3 * s + 2];
      tot += 1.f - 2.f * n / fmaxf(d, EPS);
    }
    out[0] = tot;
  }
}

extern "C" void kernel_launch(void* const* d_in, const int* in_sizes, int n_in,
                              void* d_out, int out_size, void* d_ws, size_t ws_size,
                              hipStream_t stream) {
  (void)in_sizes; (void)n_in; (void)out_size; (void)ws_size;
  const float* input   = (const float*)d_in[0];
  const int*   target  = (const int*)d_in[1];
  const int*   offsets = (const int*)d_in[2];
  float*       out     = (float*)d_out;
  float*       acc     = (float*)d_ws;   // 24 floats of scratch

  zero_acc<<<1, 32, 0, stream>>>(acc);

  dim3 grid(256, B);                      // 256 pixel-chunks x 4 batch images
  affinity_partials<<<grid, 256, 0, stream>>>(input, target, offsets, acc);

  finalize<<<1, 32, 0, stream>>>(acc, out);
}